// GNN_60885456388842
// MI455X (gfx1250) — compile-verified
//
#include <hip/hip_runtime.h>

typedef float v2f __attribute__((ext_vector_type(2)));
typedef float v8f __attribute__((ext_vector_type(8)));

#define HDIM 64
#define GBATCH 64
#define NCLS 2

// ---------------- degree / normalization ----------------
__global__ __launch_bounds__(256) void k_deg_init(float* __restrict__ deg, int n) {
  int i = blockIdx.x * blockDim.x + threadIdx.x;
  if (i < n) deg[i] = 1.0f;  // self-loop contribution
}

__global__ __launch_bounds__(256) void k_deg_edges(const int* __restrict__ dst,
                                                   float* __restrict__ deg, int e) {
  int i = blockIdx.x * blockDim.x + threadIdx.x;
  if (i < e) atomicAdd(&deg[dst[i]], 1.0f);
}

__global__ __launch_bounds__(256) void k_deg_rsqrt(float* __restrict__ deg, int n) {
  int i = blockIdx.x * blockDim.x + threadIdx.x;
  if (i < n) deg[i] = rsqrtf(deg[i]);
}

// ---------------- dense transform: O[nrows,64] = A[nrows,64] @ W[64,64] ----------------
// One wave computes one 16x16 tile of O via V_WMMA_F32_16X16X4_F32, K looped 64/4 = 16 steps.
// A-operand layout (16x4 f32): lanes 0-15 hold {K=k0,k0+1}, lanes 16-31 hold {K=k0+2,k0+3},
// row M = lane&15.  B-operand (4x16): VGPR0/1 = rows K=k0(+2*half), k0+1(+2*half), col N = lane&15.
// D (16x16 f32): VGPR i = row M = i + 8*half, col N = lane&15.
__global__ __launch_bounds__(256) void k_gemm64_wmma(const float* __restrict__ A,
                                                     const float* __restrict__ W,
                                                     float* __restrict__ O, int nrows) {
  const int wave = threadIdx.x >> 5;
  const int lane = threadIdx.x & 31;
  const int tile = blockIdx.x * 8 + wave;       // wave-uniform
  const int mtile = tile >> 2;                  // 64/16 = 4 column tiles
  const int ntile = tile & 3;
  if ((mtile << 4) >= nrows) return;            // uniform across the wave: EXEC stays all-1s
  const int m0 = mtile << 4;
  const int n0 = ntile << 4;
  const int half = lane >> 4;
  const int r = lane & 15;

  const float* arow = A + (size_t)(m0 + r) * HDIM + 2 * half;       // -> K = k0 + 2*half
  const float* bcol = W + (size_t)(2 * half) * HDIM + n0 + r;       // -> row K = k0 + 2*half

  v8f acc = {};
#pragma unroll
  for (int k0 = 0; k0 < HDIM; k0 += 4) {
    v2f a, b;
    a.x = arow[k0 + 0];
    a.y = arow[k0 + 1];
    b.x = bcol[(size_t)(k0 + 0) * HDIM];
    b.y = bcol[(size_t)(k0 + 1) * HDIM];
    acc = __builtin_amdgcn_wmma_f32_16x16x4_f32(false, a, false, b, (short)0, acc,
                                                false, false);
  }

  float* orow = O + (size_t)(m0 + 8 * half) * HDIM + n0 + r;
#pragma unroll
  for (int i = 0; i < 8; ++i) orow[(size_t)i * HDIM] = acc[i];
}

// ---------------- self-loop + bias: out[i,f] = hW[i,f]*inv[i]^2 + b[f] ----------------
__global__ __launch_bounds__(256) void k_self_bias(const float* __restrict__ hw,
                                                   const float* __restrict__ inv,
                                                   const float* __restrict__ b,
                                                   float* __restrict__ out, int n) {
  long long gid = (long long)blockIdx.x * blockDim.x + threadIdx.x;
  if (gid >= (long long)n * HDIM) return;
  int i = (int)(gid >> 6);
  int f = (int)(gid & 63);
  float w = inv[i];
  out[gid] = hw[gid] * (w * w) + b[f];
}

// ---------------- edge scatter: out[dst] += hW[src] * (inv[src]*inv[dst]) ----------------
// One lane handles 2 features (float2 gather + 2 float atomic adds); 32 lanes cover H=64.
__global__ __launch_bounds__(256) void k_edge_msg(const int* __restrict__ src,
                                                  const int* __restrict__ dst,
                                                  const float* __restrict__ inv,
                                                  const float* __restrict__ hw,
                                                  float* __restrict__ out, int e) {
  long long gid = (long long)blockIdx.x * blockDim.x + threadIdx.x;
  int eidx = (int)(gid >> 5);
  if (eidx >= e) return;
  int j = (int)(gid & 31);
  int s = src[eidx];
  int d = dst[eidx];
  float w = inv[s] * inv[d];
  v2f v = *(const v2f*)(hw + (size_t)s * HDIM + 2 * j);
  float* op = out + (size_t)d * HDIM + 2 * j;
  atomicAdd(op + 0, v.x * w);
  atomicAdd(op + 1, v.y * w);
}

// ---------------- elementwise relu (optionally out-of-place) ----------------
__global__ __launch_bounds__(256) void k_relu(const float* __restrict__ in,
                                              float* __restrict__ out, long long n) {
  long long gid = (long long)blockIdx.x * blockDim.x + threadIdx.x;
  if (gid < n) out[gid] = fmaxf(in[gid], 0.0f);
}

// ---------------- pooling ----------------
__global__ __launch_bounds__(256) void k_pool_zero(float* __restrict__ pooled,
                                                   float* __restrict__ counts) {
  int i = blockIdx.x * blockDim.x + threadIdx.x;
  if (i < GBATCH * HDIM) pooled[i] = 0.0f;
  if (i < GBATCH) counts[i] = 0.0f;
}

__global__ __launch_bounds__(256) void k_pool_accum(const float* __restrict__ h,
                                                    const int* __restrict__ batch,
                                                    float* __restrict__ pooled,
                                                    float* __restrict__ counts, int n) {
  long long gid = (long long)blockIdx.x * blockDim.x + threadIdx.x;
  if (gid >= (long long)n * HDIM) return;
  int i = (int)(gid >> 6);
  int f = (int)(gid & 63);
  int g = batch[i];
  atomicAdd(&pooled[(size_t)g * HDIM + f], h[gid]);
  if (f == 0) atomicAdd(&counts[g], 1.0f);
}

__global__ void k_head(const float* __restrict__ pooled, const float* __restrict__ counts,
                       const float* __restrict__ linW, const float* __restrict__ linb,
                       float* __restrict__ out) {
  int t = threadIdx.x;
  if (t >= GBATCH * NCLS) return;
  int g = t / NCLS;
  int c = t % NCLS;
  float invc = 1.0f / fmaxf(counts[g], 1.0f);
  float acc = linb[c];
#pragma unroll 8
  for (int k = 0; k < HDIM; ++k) acc += pooled[g * HDIM + k] * invc * linW[k * NCLS + c];
  out[t] = acc;
}

extern "C" void kernel_launch(void* const* d_in, const int* in_sizes, int n_in,
                              void* d_out, int out_size, void* d_ws, size_t ws_size,
                              hipStream_t stream) {
  const float* x    = (const float*)d_in[0];
  const int*   ei   = (const int*)d_in[1];    // [2,E]: row0 = src, row1 = dst
  const int*   batch= (const int*)d_in[2];
  const float* W1 = (const float*)d_in[3];  const float* b1 = (const float*)d_in[4];
  const float* W2 = (const float*)d_in[5];  const float* b2 = (const float*)d_in[6];
  const float* W3 = (const float*)d_in[7];  const float* b3 = (const float*)d_in[8];
  const float* linW = (const float*)d_in[9]; const float* linb = (const float*)d_in[10];

  const int N = in_sizes[0] / HDIM;   // 100000
  const int E = in_sizes[1] / 2;      // 800000
  const int* src = ei;
  const int* dst = ei + E;

  // workspace layout (floats)
  float* ws = (float*)d_ws;
  size_t Npad = ((size_t)N + 63) & ~(size_t)63;
  size_t NH = (size_t)N * HDIM;
  float* inv    = ws;                 // N  (degree -> rsqrt, in place)
  float* tA     = ws + Npad;          // N*64  transform output hW
  float* tB     = tA + NH;            // N*64  aggregation accumulator
  float* pooled = tB + NH;            // 64*64
  float* counts = pooled + (size_t)GBATCH * HDIM;  // 64

  float* out = (float*)d_out;         // [64,2] head output
  float* emb = out + GBATCH * NCLS;   // [N,64] embeddings (= relu(layer2)), also layer-3 input

  dim3 blk(256);
  int bN  = (N + 255) / 256;
  int bE  = (E + 255) / 256;
  int bNH = (int)((NH + 255) / 256);
  int tiles = (N / 16) * 4;           // N divisible by 16
  int bT  = (tiles + 7) / 8;          // 8 waves per block, one 16x16 tile per wave
  int bM  = (int)(((long long)E * 32 + 255) / 256);

  // normalization
  k_deg_init <<<bN, blk, 0, stream>>>(inv, N);
  k_deg_edges<<<bE, blk, 0, stream>>>(dst, inv, E);
  k_deg_rsqrt<<<bN, blk, 0, stream>>>(inv, N);

  // layer 1: h1 = relu(Agg(x @ W1) + b1)
  k_gemm64_wmma<<<bT, blk, 0, stream>>>(x, W1, tA, N);
  k_self_bias  <<<bNH, blk, 0, stream>>>(tA, inv, b1, tB, N);
  k_edge_msg   <<<bM, blk, 0, stream>>>(src, dst, inv, tA, tB, E);
  k_relu       <<<bNH, blk, 0, stream>>>(tB, tB, (long long)NH);

  // layer 2: emb = relu(Agg(h1 @ W2) + b2)   (written straight into d_out embeddings slice)
  k_gemm64_wmma<<<bT, blk, 0, stream>>>(tB, W2, tA, N);
  k_self_bias  <<<bNH, blk, 0, stream>>>(tA, inv, b2, tB, N);
  k_edge_msg   <<<bM, blk, 0, stream>>>(src, dst, inv, tA, tB, E);
  k_relu       <<<bNH, blk, 0, stream>>>(tB, emb, (long long)NH);

  // layer 3: h3 = Agg(emb @ W3) + b3   (no relu)
  k_gemm64_wmma<<<bT, blk, 0, stream>>>(emb, W3, tA, N);
  k_self_bias  <<<bNH, blk, 0, stream>>>(tA, inv, b3, tB, N);
  k_edge_msg   <<<bM, blk, 0, stream>>>(src, dst, inv, tA, tB, E);

  // global mean pool + linear head
  k_pool_zero <<<(GBATCH * HDIM + 255) / 256, blk, 0, stream>>>(pooled, counts);
  k_pool_accum<<<bNH, blk, 0, stream>>>(tB, batch, pooled, counts, N);
  k_head      <<<1, 128, 0, stream>>>(pooled, counts, linW, linb, out);
}